// Affine_Transformer_6906307412225
// MI455X (gfx1250) — compile-verified
//
#include <hip/hip_runtime.h>

// Affine bound propagation (CROWN-style), D=2048, L=8 sequential steps.
// Per step: 4 fp32 GEMMs (2048^3) on the WMMA f32 16x16x4 pipe + 2 matvecs.
// GEMM staging uses gfx1250 async global->LDS DMA (ASYNCcnt) with a
// double-buffered LDS pipeline so HBM latency overlaps the XDL WMMA pipe.
// Carry ping-pongs through d_ws (needs 4*D*D + 4*D floats ~= 64 MB).

#define D 2048
#define LSTEPS 8
#define BM 128
#define BN 128
#define BK 32
#define NCHUNK (D / BK)

typedef __attribute__((ext_vector_type(2))) float v2f;
typedef __attribute__((ext_vector_type(8))) float v8f;

// Dst_tile = relu(S) @ Bp + min(S,0) @ Bn  for a 128x128 tile.
// z==0: S=Al, Bp=dAl, Bn=dAu -> DstL ; z==1: S=Au, Bp=dAu, Bn=dAl -> DstU
__global__ __launch_bounds__(256) void ibp_gemm(
    const float* __restrict__ SrcL, const float* __restrict__ SrcU,
    const float* __restrict__ dAl,  const float* __restrict__ dAu,
    float* __restrict__ DstL,       float* __restrict__ DstU)
{
  const float* S;  const float* Bp; const float* Bn; float* Dst;
  if (blockIdx.z == 0) { S = SrcL; Bp = dAl; Bn = dAu; Dst = DstL; }
  else                 { S = SrcU; Bp = dAu; Bn = dAl; Dst = DstU; }

  // double-buffered tiles (~100 KB of the 320 KB WGP LDS pool)
  __shared__ float lA [2][BM][BK + 2];   // row stride 34 floats (8B-aligned k-pairs)
  __shared__ float lBp[2][BK][BN + 4];   // row stride 132 floats
  __shared__ float lBn[2][BK][BN + 4];

  const int tid  = threadIdx.x;
  const int lane = tid & 31;
  const int w    = tid >> 5;          // 8 waves
  const int i0   = blockIdx.y * BM;
  const int j0   = blockIdx.x * BN;
  const int wr   = (w & 3) * 32;      // wave row offset in tile
  const int wc   = (w >> 2) * 64;     // wave col offset in tile
  const int lm   = lane & 15;         // M / N index within 16-tile
  const int kh   = (lane >> 4) << 1;  // K pair select (0 or 2)

  // per-thread staging coordinates (10 x b128 async DMA per thread per chunk)
  const int ar  = tid >> 1;                 // A row 0..127
  const int ac4 = (tid & 1) << 4;           // A col 0 or 16 (2 x float4 pairs)
  const int br  = tid >> 3;                 // B row 0..31
  const int bc4 = (tid & 7) << 4;           // B col {0,16,...,112}

  v8f acc[2][4];
#pragma unroll
  for (int rt = 0; rt < 2; ++rt)
#pragma unroll
    for (int ct = 0; ct < 4; ++ct)
      acc[rt][ct] = {};

  // ---- async DMA staging of one K-chunk into LDS buffer `buf` ----
  auto stage_async = [&](int buf, int k0) {
    // A tile: BM x BK floats; thread moves 2 contiguous float4 of one row
#pragma unroll
    for (int q = 0; q < 2; ++q) {
      unsigned loff = (unsigned)(uintptr_t)(&lA[buf][ar][ac4 + q * 4]);
      unsigned long long g =
          (unsigned long long)(uintptr_t)(S + (size_t)(i0 + ar) * D + k0 + ac4 + q * 4);
      asm volatile("global_load_async_to_lds_b128 %0, %1, off"
                   :: "v"(loff), "v"(g) : "memory");
    }
    // B tiles: BK x BN floats each; thread moves 4 float4 of one row, x2 mats
#pragma unroll
    for (int q = 0; q < 4; ++q) {
      unsigned lp = (unsigned)(uintptr_t)(&lBp[buf][br][bc4 + q * 4]);
      unsigned ln = (unsigned)(uintptr_t)(&lBn[buf][br][bc4 + q * 4]);
      unsigned long long gp =
          (unsigned long long)(uintptr_t)(Bp + (size_t)(k0 + br) * D + j0 + bc4 + q * 4);
      unsigned long long gn =
          (unsigned long long)(uintptr_t)(Bn + (size_t)(k0 + br) * D + j0 + bc4 + q * 4);
      asm volatile("global_load_async_to_lds_b128 %0, %1, off"
                   :: "v"(lp), "v"(gp) : "memory");
      asm volatile("global_load_async_to_lds_b128 %0, %1, off"
                   :: "v"(ln), "v"(gn) : "memory");
    }
  };

  stage_async(0, 0);
  asm volatile("s_wait_asynccnt 0x0" ::: "memory");
  __syncthreads();

  for (int c = 0; c < NCHUNK; ++c) {
    const int cur = c & 1;
    // kick off next chunk's DMA before computing -> overlaps WMMA pipe
    if (c + 1 < NCHUNK) stage_async(cur ^ 1, (c + 1) * BK);

    // ---- compute: 8 K-steps of 4, 16 WMMA per K-step per wave ----
#pragma unroll
    for (int kk = 0; kk < BK; kk += 4) {
      const int kl = kk + kh;
      v2f ap[2], an[2];
#pragma unroll
      for (int rt = 0; rt < 2; ++rt) {
        const int m = wr + rt * 16 + lm;
        float a0 = lA[cur][m][kl], a1 = lA[cur][m][kl + 1];
        ap[rt].x = fmaxf(a0, 0.f); ap[rt].y = fmaxf(a1, 0.f);
        an[rt].x = fminf(a0, 0.f); an[rt].y = fminf(a1, 0.f);
      }
#pragma unroll
      for (int ct = 0; ct < 4; ++ct) {
        const int n = wc + ct * 16 + lm;
        v2f bp, bn;
        bp.x = lBp[cur][kl][n]; bp.y = lBp[cur][kl + 1][n];
        bn.x = lBn[cur][kl][n]; bn.y = lBn[cur][kl + 1][n];
#pragma unroll
        for (int rt = 0; rt < 2; ++rt) {
          acc[rt][ct] = __builtin_amdgcn_wmma_f32_16x16x4_f32(
              false, ap[rt], false, bp, (short)0, acc[rt][ct], false, false);
          acc[rt][ct] = __builtin_amdgcn_wmma_f32_16x16x4_f32(
              false, an[rt], false, bn, (short)0, acc[rt][ct], false, false);
        }
      }
    }

    // my DMA done, then all waves' DMA visible + compute done for this buffer
    asm volatile("s_wait_asynccnt 0x0" ::: "memory");
    __syncthreads();
  }

  // ---- write back: C/D layout -> VGPR r holds M=r (lanes 0-15) / M=r+8 ----
#pragma unroll
  for (int rt = 0; rt < 2; ++rt)
#pragma unroll
    for (int ct = 0; ct < 4; ++ct)
#pragma unroll
      for (int r = 0; r < 8; ++r) {
        int row = i0 + wr + rt * 16 + ((lane >> 4) << 3) + r;
        int col = j0 + wc + ct * 16 + lm;
        Dst[(size_t)row * D + col] = acc[rt][ct][r];
      }
}

// dst[row] = relu(S[row,:]) . vpos + min(S[row,:],0) . vneg + bias[row]
// one wave per row (wave32), 8 rows per 256-thread block
__global__ __launch_bounds__(256) void ibp_matvec(
    const float* __restrict__ S, const float* __restrict__ vpos,
    const float* __restrict__ vneg, const float* __restrict__ bias,
    float* __restrict__ dst)
{
  const int lane = threadIdx.x & 31;
  const int w    = threadIdx.x >> 5;
  const int row  = blockIdx.x * 8 + w;
  const float* srow = S + (size_t)row * D;

  float acc = 0.f;
  for (int k = lane * 4; k < D; k += 32 * 4) {
    float4 a = *(const float4*)(srow + k);
    float4 p = *(const float4*)(vpos + k);
    float4 n = *(const float4*)(vneg + k);
    acc += fmaxf(a.x, 0.f) * p.x + fminf(a.x, 0.f) * n.x;
    acc += fmaxf(a.y, 0.f) * p.y + fminf(a.y, 0.f) * n.y;
    acc += fmaxf(a.z, 0.f) * p.z + fminf(a.z, 0.f) * n.z;
    acc += fmaxf(a.w, 0.f) * p.w + fminf(a.w, 0.f) * n.w;
  }
#pragma unroll
  for (int off = 16; off > 0; off >>= 1)
    acc += __shfl_down(acc, off, 32);
  if (lane == 0) dst[row] = acc + bias[row];
}

extern "C" void kernel_launch(void* const* d_in, const int* in_sizes, int n_in,
                              void* d_out, int out_size, void* d_ws, size_t ws_size,
                              hipStream_t stream) {
  const float* A        = (const float*)d_in[0];
  const float* b        = (const float*)d_in[1];
  const float* hAl      = (const float*)d_in[2];
  const float* hAu      = (const float*)d_in[3];
  const float* hbl      = (const float*)d_in[4];
  const float* hbu      = (const float*)d_in[5];
  const float* lower_in = (const float*)d_in[6];
  const float* upper_in = (const float*)d_in[7];
  float* out = (float*)d_out;

  const size_t mat = (size_t)D * D;
  float* ws = (float*)d_ws;
  float* AlBuf[2] = { ws + 0 * mat, ws + 1 * mat };
  float* AuBuf[2] = { ws + 2 * mat, ws + 3 * mat };
  float* blBuf[2] = { ws + 4 * mat,           ws + 4 * mat + D };
  float* buBuf[2] = { ws + 4 * mat + 2 * D,   ws + 4 * mat + 3 * D };

  dim3 blk(256);
  dim3 gGemm(D / BN, D / BM, 2);  // 16 x 16 x 2
  dim3 gMv(D / 8);

  for (int s = 0; s < LSTEPS; ++s) {
    const float* sL  = (s == 0) ? A : AlBuf[(s - 1) & 1];
    const float* sU  = (s == 0) ? A : AuBuf[(s - 1) & 1];
    const float* sbl = (s == 0) ? b : blBuf[(s - 1) & 1];
    const float* sbu = (s == 0) ? b : buBuf[(s - 1) & 1];
    const float* dAl = hAl + (size_t)(LSTEPS - 1 - s) * mat;
    const float* dAu = hAu + (size_t)(LSTEPS - 1 - s) * mat;
    const float* dbl = hbl + (size_t)(LSTEPS - 1 - s) * D;
    const float* dbu = hbu + (size_t)(LSTEPS - 1 - s) * D;

    ibp_gemm<<<gGemm, blk, 0, stream>>>(sL, sU, dAl, dAu, AlBuf[s & 1], AuBuf[s & 1]);
    ibp_matvec<<<gMv, blk, 0, stream>>>(sL, dbl, dbu, sbl, blBuf[s & 1]);
    ibp_matvec<<<gMv, blk, 0, stream>>>(sU, dbu, dbl, sbu, buBuf[s & 1]);
  }

  const int f = (LSTEPS - 1) & 1;
  ibp_matvec<<<gMv, blk, 0, stream>>>(AlBuf[f], lower_in, upper_in, blBuf[f], out);
  ibp_matvec<<<gMv, blk, 0, stream>>>(AuBuf[f], upper_in, lower_in, buBuf[f], out + D);
}